// GCN_4011499454825
// MI455X (gfx1250) — compile-verified
//
#include <hip/hip_runtime.h>
#include <hip/hip_bf16.h>
#include <stdint.h>

// ---------------- problem constants (match reference) ----------------
#define Nn 100000
#define Ee 1600000
#define Gg 512
#define Hh 128
#define Cc 10
#define CH 512                      // scan chunk size
#define NCH ((Nn + CH - 1) / CH)    // 196 chunks

typedef float v2f __attribute__((ext_vector_type(2)));
typedef float v8f __attribute__((ext_vector_type(8)));

// ---------------- utility fills ----------------
__global__ void fill_f32_kernel(float* p, float v, int n) {
    int i = blockIdx.x * blockDim.x + threadIdx.x;
    if (i < n) p[i] = v;
}
__global__ void fill_i32_kernel(int* p, int v, int n) {
    int i = blockIdx.x * blockDim.x + threadIdx.x;
    if (i < n) p[i] = v;
}

// ---------------- degree / normalization ----------------
__global__ void deg_count_kernel(const long long* __restrict__ ei, int* __restrict__ deg) {
    int e = blockIdx.x * blockDim.x + threadIdx.x;
    if (e >= Ee) return;
    atomicAdd(&deg[(int)ei[(size_t)Ee + e]], 1);
}
__global__ void dis_kernel(const int* __restrict__ deg, float* __restrict__ dis) {
    int i = blockIdx.x * blockDim.x + threadIdx.x;
    if (i >= Nn) return;
    dis[i] = rsqrtf((float)deg[i] + 1.0f);   // +1 self loop => deg > 0 always
}

// ---------------- CSR build: chunk reduce -> chunk scan -> node offsets ----
__global__ void chunk_sum_kernel(const int* __restrict__ deg, int* __restrict__ csum) {
    __shared__ int sh[CH];
    const int t = threadIdx.x, g = blockIdx.x * CH + t;
    sh[t] = (g < Nn) ? deg[g] : 0;
    __syncthreads();
    for (int off = CH / 2; off > 0; off >>= 1) {
        if (t < off) sh[t] += sh[t + off];
        __syncthreads();
    }
    if (t == 0) csum[blockIdx.x] = sh[0];
}

__global__ void chunk_scan_kernel(int* __restrict__ csum) {  // single block, CH threads
    __shared__ int sh[CH];
    const int t = threadIdx.x;
    const int own = (t < NCH) ? csum[t] : 0;
    sh[t] = own;
    __syncthreads();
    for (int off = 1; off < CH; off <<= 1) {               // Hillis-Steele inclusive
        int v = (t >= off) ? sh[t - off] : 0;
        __syncthreads();
        sh[t] += v;
        __syncthreads();
    }
    if (t < NCH) csum[t] = sh[t] - own;                    // exclusive
}

__global__ void node_offset_kernel(const int* __restrict__ deg, const int* __restrict__ csum,
                                   int* __restrict__ rowstart, int* __restrict__ cursor) {
    __shared__ int sh[CH];
    const int t = threadIdx.x, g = blockIdx.x * CH + t;
    const int own = (g < Nn) ? deg[g] : 0;
    sh[t] = own;
    __syncthreads();
    for (int off = 1; off < CH; off <<= 1) {
        int v = (t >= off) ? sh[t - off] : 0;
        __syncthreads();
        sh[t] += v;
        __syncthreads();
    }
    if (g < Nn) {
        const int ex = csum[blockIdx.x] + sh[t] - own;     // exclusive prefix
        rowstart[g] = ex;
        cursor[g]   = ex;
    }
}

__global__ void csr_fill_kernel(const long long* __restrict__ ei,
                                const float* __restrict__ edgemask,
                                const float* __restrict__ dis,
                                int* __restrict__ cursor, int* __restrict__ csr_src,
                                float* __restrict__ csr_wm, float* __restrict__ csr_wn) {
    int e = blockIdx.x * blockDim.x + threadIdx.x;
    if (e >= Ee) return;
    const int s = (int)ei[e];
    const int d = (int)ei[(size_t)Ee + e];
    const int pos = atomicAdd(&cursor[d], 1);
    csr_src[pos] = s;
    csr_wm[pos]  = edgemask[e];
    csr_wn[pos]  = dis[s] * dis[d];
}

// ---------------- SimpleConv('add') + supernode overwrite (wave per node) --
__global__ void simpleconv_kernel(const float* __restrict__ x,
                                  const unsigned char* __restrict__ mask,
                                  const int* __restrict__ rowstart,
                                  const int* __restrict__ csr_src,
                                  const float* __restrict__ csr_wm,
                                  float* __restrict__ Xout) {
    const int i = blockIdx.x * 8 + (threadIdx.x >> 5);
    if (i >= Nn) return;
    const int col = (threadIdx.x & 31) * 4;
    float4 acc;
    if (mask[i]) {
        acc = make_float4(0.f, 0.f, 0.f, 0.f);
        const int end = rowstart[i + 1];
        for (int e = rowstart[i]; e < end; ++e) {
            const float w = csr_wm[e];
            if (w != 0.0f) {
                const float4 v = *(const float4*)(x + (size_t)csr_src[e] * Hh + col);
                acc.x = fmaf(w, v.x, acc.x); acc.y = fmaf(w, v.y, acc.y);
                acc.z = fmaf(w, v.z, acc.z); acc.w = fmaf(w, v.w, acc.w);
            }
        }
    } else {
        acc = *(const float4*)(x + (size_t)i * Hh + col);
    }
    *(float4*)(Xout + (size_t)i * Hh + col) = acc;
}

// ---------------- GCN aggregation, fused self-loop + bias + relu -----------
template <int RELU>
__global__ void gcn_aggregate_kernel(const float* __restrict__ Hfeat,
                                     const int* __restrict__ rowstart,
                                     const int* __restrict__ csr_src,
                                     const float* __restrict__ csr_wn,
                                     const float* __restrict__ dis,
                                     const float* __restrict__ bias,
                                     float* __restrict__ Xout) {
    const int i = blockIdx.x * 8 + (threadIdx.x >> 5);
    if (i >= Nn) return;
    const int col = (threadIdx.x & 31) * 4;
    float4 acc = make_float4(0.f, 0.f, 0.f, 0.f);
    const int end = rowstart[i + 1];
    for (int e = rowstart[i]; e < end; ++e) {
        const float w  = csr_wn[e];
        const float4 v = *(const float4*)(Hfeat + (size_t)csr_src[e] * Hh + col);
        acc.x = fmaf(w, v.x, acc.x); acc.y = fmaf(w, v.y, acc.y);
        acc.z = fmaf(w, v.z, acc.z); acc.w = fmaf(w, v.w, acc.w);
    }
    const float di = dis[i], ws = di * di;                 // self-loop weight
    const float4 hv = *(const float4*)(Hfeat + (size_t)i * Hh + col);
    acc.x = fmaf(ws, hv.x, acc.x); acc.y = fmaf(ws, hv.y, acc.y);
    acc.z = fmaf(ws, hv.z, acc.z); acc.w = fmaf(ws, hv.w, acc.w);
    const float4 bb = *(const float4*)(bias + col);
    acc.x += bb.x; acc.y += bb.y; acc.z += bb.z; acc.w += bb.w;
    if (RELU) {
        acc.x = fmaxf(acc.x, 0.f); acc.y = fmaxf(acc.y, 0.f);
        acc.z = fmaxf(acc.z, 0.f); acc.w = fmaxf(acc.w, 0.f);
    }
    *(float4*)(Xout + (size_t)i * Hh + col) = acc;
}

// ---------------- WMMA fp32 GEMM: Out[M,128] = A[M,128] @ W[128,128] -------
// W staged into LDS once per block via gfx1250 async-to-LDS (inline asm,
// tracked by ASYNCcnt); one 16x16 tile per wave via V_WMMA_F32_16X16X4_F32;
// block = 8 waves = one 16-row output stripe.
template <int RELU, int HAS_BIAS>
__global__ void wmma_gemm128_kernel(const float* __restrict__ A,
                                    const float* __restrict__ W,
                                    const float* __restrict__ bias,
                                    float* __restrict__ Out) {
    __shared__ float wlds[Hh * Hh];                        // 64 KB
    #pragma unroll
    for (int it = 0; it < 16; ++it) {                      // 256 thr x 16B x 16 = 64 KB
        const int idx = (it * 256 + threadIdx.x) * 4;
        // Low 32 bits of a generic LDS pointer = workgroup-relative LDS offset
        const uint32_t lds_off = (uint32_t)(uintptr_t)(&wlds[idx]);
        const float* gp = W + idx;
        asm volatile("global_load_async_to_lds_b128 %0, %1, off"
                     :: "v"(lds_off), "v"(gp)
                     : "memory");
    }
    asm volatile("s_wait_asynccnt 0x0" ::: "memory");      // async fills landed
    __syncthreads();

    const int m0   = blockIdx.x * 16;
    const int n0   = (threadIdx.x >> 5) * 16;              // wave -> column tile
    const int l    = threadIdx.x & 31;
    const int mn   = l & 15;
    const int half = l >> 4;                               // K sub-offset 0 or 2
    v8f acc = {};
    const float* arow = A + (size_t)(m0 + mn) * Hh;
    #pragma unroll 4
    for (int k0 = 0; k0 < Hh; k0 += 4) {
        const int kk = k0 + 2 * half;
        v2f a, b;
        a.x = arow[kk];
        a.y = arow[kk + 1];
        b.x = wlds[kk * Hh + n0 + mn];
        b.y = wlds[(kk + 1) * Hh + n0 + mn];
        acc = __builtin_amdgcn_wmma_f32_16x16x4_f32(
            false, a, false, b, (short)0, acc, false, false);
    }
    #pragma unroll
    for (int r = 0; r < 8; ++r) {
        const int row = m0 + r + 8 * half;
        const int col = n0 + mn;
        float v = acc[r];
        if (HAS_BIAS) v += bias[col];
        if (RELU)     v = fmaxf(v, 0.0f);
        Out[(size_t)row * Hh + col] = v;
    }
}

// ---------------- global add pool ----------------
__global__ void pool_kernel(const float* __restrict__ X,
                            const long long* __restrict__ batch,
                            float* __restrict__ g, int n_elems) {
    int idx = blockIdx.x * blockDim.x + threadIdx.x;
    if (idx >= n_elems) return;
    const int i = idx >> 7, c = idx & 127;
    atomicAdd(&g[(size_t)batch[i] * Hh + c], X[idx]);
}

// ---------------- final tiny GEMM: out[512,10] = g1 @ Wm2 + bm2 ------------
__global__ void mlp2_kernel(const float* __restrict__ g1, const float* __restrict__ Wm2,
                            const float* __restrict__ bm2, float* __restrict__ out) {
    int idx = blockIdx.x * blockDim.x + threadIdx.x;
    if (idx >= Gg * Cc) return;
    const int gi = idx / Cc, c = idx % Cc;
    const float* row = g1 + (size_t)gi * Hh;
    float acc = bm2[c];
    #pragma unroll 8
    for (int k = 0; k < Hh; ++k) acc = fmaf(row[k], Wm2[k * Cc + c], acc);
    out[idx] = acc;
}

// =======================================================================
extern "C" void kernel_launch(void* const* d_in, const int* in_sizes, int n_in,
                              void* d_out, int out_size, void* d_ws, size_t ws_size,
                              hipStream_t stream) {
    const float*         x        = (const float*)d_in[0];
    const long long*     ei       = (const long long*)d_in[1];
    const unsigned char* supmask  = (const unsigned char*)d_in[2];
    const float*         edgemask = (const float*)d_in[3];
    const long long*     batch    = (const long long*)d_in[4];
    const float* W1  = (const float*)d_in[5];   const float* b1  = (const float*)d_in[6];
    const float* W2  = (const float*)d_in[7];   const float* b2  = (const float*)d_in[8];
    const float* W3  = (const float*)d_in[9];   const float* b3  = (const float*)d_in[10];
    const float* Wm1 = (const float*)d_in[11];  const float* bm1 = (const float*)d_in[12];
    const float* Wm2 = (const float*)d_in[13];  const float* bm2 = (const float*)d_in[14];
    float* out = (float*)d_out;

    // ---- workspace carve-up (16B-aligned slices) ----
    const size_t NH = (size_t)Nn * Hh;
    float* Xb     = (float*)d_ws;               // [N,128]
    float* Hb     = Xb + NH;                    // [N,128]
    float* dis    = Hb + NH;                    // [N]
    float* csr_wm = dis + Nn;                   // [E]
    float* csr_wn = csr_wm + Ee;                // [E]
    float* gbuf   = csr_wn + Ee;                // [512,128]
    float* g1     = gbuf + (size_t)Gg * Hh;     // [512,128]
    int* deg      = (int*)(g1 + (size_t)Gg * Hh);  // [N]
    int* rowstart = deg + Nn;                   // [N+1] (padded to N+4)
    int* cursor   = rowstart + Nn + 4;          // [N]
    int* csum     = cursor + Nn;                // [NCH] (pad 512)
    int* csr_src  = csum + CH;                  // [E]

    const int T   = 256;
    const int nNH = (int)NH;
    const int gNH = (nNH + T - 1) / T;
    const int gN  = (Nn + T - 1) / T;
    const int gE  = (Ee + T - 1) / T;
    const int gNd = (Nn + 7) / 8;               // 8 node-waves per block
    const int gRow = Nn / 16;                   // 6250 WMMA row stripes

    // 1) degree + D^-1/2
    fill_i32_kernel<<<gN, T, 0, stream>>>(deg, 0, Nn);
    deg_count_kernel<<<gE, T, 0, stream>>>(ei, deg);
    dis_kernel<<<gN, T, 0, stream>>>(deg, dis);

    // 2) CSR by destination: scan degrees, fill buckets (once per call)
    chunk_sum_kernel<<<NCH, CH, 0, stream>>>(deg, csum);
    chunk_scan_kernel<<<1, CH, 0, stream>>>(csum);
    node_offset_kernel<<<NCH, CH, 0, stream>>>(deg, csum, rowstart, cursor);
    fill_i32_kernel<<<1, 1, 0, stream>>>(rowstart + Nn, Ee, 1);
    csr_fill_kernel<<<gE, T, 0, stream>>>(ei, edgemask, dis, cursor, csr_src, csr_wm, csr_wn);

    // 3) SimpleConv('add') with edge_mask, supernode rows only
    simpleconv_kernel<<<gNd, T, 0, stream>>>(x, supmask, rowstart, csr_src, csr_wm, Xb);

    // 4) three GCN conv layers (gather aggregation, no atomics)
    wmma_gemm128_kernel<0, 0><<<gRow, T, 0, stream>>>(Xb, W1, nullptr, Hb);
    gcn_aggregate_kernel<1><<<gNd, T, 0, stream>>>(Hb, rowstart, csr_src, csr_wn, dis, b1, Xb);
    wmma_gemm128_kernel<0, 0><<<gRow, T, 0, stream>>>(Xb, W2, nullptr, Hb);
    gcn_aggregate_kernel<1><<<gNd, T, 0, stream>>>(Hb, rowstart, csr_src, csr_wn, dis, b2, Xb);
    wmma_gemm128_kernel<0, 0><<<gRow, T, 0, stream>>>(Xb, W3, nullptr, Hb);
    gcn_aggregate_kernel<0><<<gNd, T, 0, stream>>>(Hb, rowstart, csr_src, csr_wn, dis, b3, Xb);

    // 5) global add pool -> [512,128]
    const int nGH = Gg * Hh;
    fill_f32_kernel<<<(nGH + T - 1) / T, T, 0, stream>>>(gbuf, 0.0f, nGH);
    pool_kernel<<<gNH, T, 0, stream>>>(Xb, batch, gbuf, nNH);

    // 6) MLP: relu(g@Wm1+bm1) @ Wm2 + bm2
    wmma_gemm128_kernel<1, 1><<<Gg / 16, T, 0, stream>>>(gbuf, Wm1, bm1, g1);
    mlp2_kernel<<<(Gg * Cc + T - 1) / T, T, 0, stream>>>(g1, Wm2, bm2, out);
}